// MultiScalePeriodicPatchEmbedding_19430432047493
// MI455X (gfx1250) — compile-verified
//
#include <hip/hip_runtime.h>
#include <math.h>

// ---------------------------------------------------------------------------
// MultiScalePeriodicPatchEmbedding for MI455X (gfx1250, wave32)
// fp32 end-to-end; matrix work on V_WMMA_F32_16X16X4_F32.
// Store-bandwidth-bound: ~217 MB of output, so fp32 WMMA is far from the
// bottleneck and preserves reference precision.
// ---------------------------------------------------------------------------

typedef __attribute__((ext_vector_type(2))) float v2f;
typedef __attribute__((ext_vector_type(8))) float v8f;

static __device__ __forceinline__ v8f wmma_f32(v2f a, v2f b, v8f c) {
  // (neg_a, A, neg_b, B, c_mod, C, reuse_a, reuse_b)
  return __builtin_amdgcn_wmma_f32_16x16x4_f32(false, a, false, b, (short)0, c,
                                               false, false);
}

#define NE 43
#define FREQ 256
#define HID 1024

// Patch sizes / patch counts derived from SEQ_LEN=512 (fixed by the problem).
__constant__ int c_P[NE] = {2,  3,  4,  5,  6,  7,  8,  9,  10, 11, 12,
                            13, 14, 15, 16, 17, 18, 19, 20, 21, 22, 23,
                            24, 25, 26, 28, 30, 32, 34, 36, 39, 42, 46,
                            51, 56, 64, 73, 85, 102,128,170,256,512};
__constant__ int c_N[NE] = {256,171,128,103,86, 74, 64, 57, 52, 47, 43,
                            40, 37, 35, 32, 31, 29, 27, 26, 25, 24, 23,
                            22, 21, 20, 19, 18, 16, 16, 15, 14, 13, 12,
                            11, 10, 8,  8,  7,  6,  4,  4,  2,  1};
__constant__ int c_cumN[NE + 1] = {
    0,   256, 427, 555, 658, 744, 818, 882, 939, 991, 1038,
    1081,1121,1158,1193,1225,1256,1285,1312,1338,1363,1387,
    1410,1432,1453,1473,1492,1510,1526,1542,1557,1571,1584,
    1596,1607,1617,1625,1633,1640,1646,1650,1654,1656,1657};

// Workspace layout (float offsets)
#define XR_OFF  0         // [64][256]
#define XI_OFF  16384     // [64][256]
#define O1R_OFF 32768     // [64][1024]
#define O1I_OFF 98304     // [64][1024]
#define MAG_OFF 163840    // [64][48]

// ---------------------------------------------------------------------------
// 1) DFT (rfft, k = 1..256) with exact integer-phase twiddles in LDS.
//    grid = 64 (one block per (b,c) row), block = 256.
// ---------------------------------------------------------------------------
__global__ __launch_bounds__(256) void dft_kernel(const float* __restrict__ x,
                                                  float* __restrict__ ws) {
  __shared__ float xrow[512];
  __shared__ float ctab[512];
  const int row = blockIdx.x;  // b*8 + c
  const int b   = row >> 3;
  const int ch  = row & 7;
  const int t   = threadIdx.x;  // 0..255
  const float w0 = 6.2831853071795864769f / 512.0f;
  ctab[t]       = cosf(w0 * (float)t);
  ctab[t + 256] = cosf(w0 * (float)(t + 256));
  xrow[t]       = x[(b * 512 + t) * 8 + ch];
  xrow[t + 256] = x[(b * 512 + t + 256) * 8 + ch];
  __syncthreads();
  const int k = t + 1;  // drop DC
  float sr = 0.0f, si = 0.0f;
  for (int tt = 0; tt < 512; ++tt) {
    const int m = (k * tt) & 511;
    const float v = xrow[tt];
    sr = fmaf(v, ctab[m], sr);                 // cos term
    si = fmaf(v, ctab[(m + 384) & 511], si);   // sin term
  }
  ws[XR_OFF + row * FREQ + t] = sr;
  ws[XI_OFF + row * FREQ + t] = -si;  // rfft imag = -sum sin
}

// ---------------------------------------------------------------------------
// 2) Complex layer 1: [64x256] @ [256x1024] (4 cross-term WMMAs / K-step),
//    bias + relu. grid = 32 blocks x 256 threads, one wave per 16x16 tile.
// ---------------------------------------------------------------------------
__global__ __launch_bounds__(256) void mlp1_kernel(float* __restrict__ ws,
                                                   const float* __restrict__ w1,
                                                   const float* __restrict__ b1) {
  const int wid  = (blockIdx.x << 3) + (threadIdx.x >> 5);  // 0..255
  const int lane = threadIdx.x & 31;
  const int mt = wid >> 6;   // 0..3
  const int nt = wid & 63;   // 0..63
  const int hl = lane >> 4;
  const int mr = lane & 15;
  const int rowA = mt * 16 + mr;
  const int ncol = nt * 16 + mr;
  const float* xr  = ws + XR_OFF;
  const float* xi  = ws + XI_OFF;
  const float* w1r = w1;
  const float* w1i = w1 + 256 * 1024;
  v8f cr = {}, ci = {};
  for (int k0 = 0; k0 < FREQ; k0 += 4) {
    const int ka = k0 + hl * 2;
    v2f ar = *(const v2f*)(xr + rowA * FREQ + ka);
    v2f ai = *(const v2f*)(xi + rowA * FREQ + ka);
    v2f br, bi;
    br.x = w1r[ka * HID + ncol];
    br.y = w1r[(ka + 1) * HID + ncol];
    bi.x = w1i[ka * HID + ncol];
    bi.y = w1i[(ka + 1) * HID + ncol];
    v2f nai = -ai;  // f32 WMMA has no A-neg modifier; negate in VALU
    cr = wmma_f32(ar, br, cr);
    cr = wmma_f32(nai, bi, cr);
    ci = wmma_f32(ai, br, ci);
    ci = wmma_f32(ar, bi, ci);
  }
  const float biasr = b1[ncol];
  const float biasi = b1[HID + ncol];
  float* o1r = ws + O1R_OFF;
  float* o1i = ws + O1I_OFF;
#pragma unroll
  for (int j = 0; j < 8; ++j) {
    const int row = mt * 16 + j + hl * 8;
    float vr = cr[j] + biasr; vr = vr > 0.0f ? vr : 0.0f;
    float vi = ci[j] + biasi; vi = vi > 0.0f ? vi : 0.0f;
    o1r[row * HID + ncol] = vr;
    o1i[row * HID + ncol] = vi;
  }
}

// ---------------------------------------------------------------------------
// 3) Complex layer 2: [64x1024] @ [1024x43] (N padded to 48), then magnitude.
//    grid = (4,3) blocks of 32 threads (one wave per tile).
// ---------------------------------------------------------------------------
__global__ __launch_bounds__(32) void mlp2_kernel(float* __restrict__ ws,
                                                  const float* __restrict__ w2,
                                                  const float* __restrict__ b2) {
  const int lane = threadIdx.x;
  const int mt = blockIdx.x;  // 0..3
  const int nt = blockIdx.y;  // 0..2
  const int hl = lane >> 4;
  const int mr = lane & 15;
  const int rowA = mt * 16 + mr;
  const int ncol = nt * 16 + mr;      // 0..47
  const bool valid = ncol < NE;
  const float* o1r = ws + O1R_OFF;
  const float* o1i = ws + O1I_OFF;
  const float* w2r = w2;
  const float* w2i = w2 + HID * NE;
  v8f cr = {}, ci = {};
  for (int k0 = 0; k0 < HID; k0 += 4) {
    const int ka = k0 + hl * 2;
    v2f ar = *(const v2f*)(o1r + rowA * HID + ka);
    v2f ai = *(const v2f*)(o1i + rowA * HID + ka);
    v2f br, bi;
    br.x = valid ? w2r[ka * NE + ncol] : 0.0f;
    br.y = valid ? w2r[(ka + 1) * NE + ncol] : 0.0f;
    bi.x = valid ? w2i[ka * NE + ncol] : 0.0f;
    bi.y = valid ? w2i[(ka + 1) * NE + ncol] : 0.0f;
    v2f nai = -ai;
    cr = wmma_f32(ar, br, cr);
    cr = wmma_f32(nai, bi, cr);
    ci = wmma_f32(ai, br, ci);
    ci = wmma_f32(ar, bi, ci);
  }
  const float biasr = valid ? b2[ncol] : 0.0f;
  const float biasi = valid ? b2[NE + ncol] : 0.0f;
  float* mag = ws + MAG_OFF;
#pragma unroll
  for (int j = 0; j < 8; ++j) {
    const int row = mt * 16 + j + hl * 8;
    const float orv = cr[j] + biasr;
    const float oiv = ci[j] + biasi;
    mag[row * 48 + ncol] = sqrtf(orv * orv + oiv * oiv);
  }
}

// ---------------------------------------------------------------------------
// 4) Mean over C, top-5, softmax, scatter to gates (tail of d_out).
// ---------------------------------------------------------------------------
__global__ __launch_bounds__(64) void gates_kernel(const float* __restrict__ ws,
                                                   float* __restrict__ gates) {
  const int b = threadIdx.x;
  if (b >= 8) return;
  const float* mag = ws + MAG_OFF;
  float w[NE];
  for (int e = 0; e < NE; ++e) {
    float s = 0.0f;
    for (int c = 0; c < 8; ++c) s += mag[(b * 8 + c) * 48 + e];
    w[e] = s * 0.125f;
  }
  int idx[5];
  float val[5];
  for (int i = 0; i < 5; ++i) {
    float best = -INFINITY; int bi = 0;
    for (int e = 0; e < NE; ++e) {
      if (w[e] > best) { best = w[e]; bi = e; }  // strict > keeps first on ties
    }
    idx[i] = bi; val[i] = best; w[bi] = -INFINITY;
  }
  const float mx = val[0];
  float sum = 0.0f, sm[5];
  for (int i = 0; i < 5; ++i) { sm[i] = expf(val[i] - mx); sum += sm[i]; }
  const float inv = 1.0f / sum;
  for (int e = 0; e < NE; ++e) gates[b * NE + e] = 0.0f;
  for (int i = 0; i < 5; ++i) gates[b * NE + idx[i]] = sm[i] * inv;
}

// ---------------------------------------------------------------------------
// 5) Patch embedding: one block per 16-row M-tile (6628 total).
//    A-tile (16 x Kpad, replication-padded) staged to LDS once; 8 waves each
//    fuse 4 N-tiles in one K-loop (1 A-frag load amortized over 4 WMMAs).
//    Gate mask folded into the store (zeros written where gate == 0).
// ---------------------------------------------------------------------------
struct ExpertPtrs { const float* w[NE]; };

__global__ __launch_bounds__(256) void embed_kernel(const float* __restrict__ x,
                                                    const float* __restrict__ gates,
                                                    float* __restrict__ out,
                                                    ExpertPtrs ptrs) {
  __shared__ float lA[16 * 512];
  __shared__ float lScale[16];

  const int mt = blockIdx.x;  // global M-tile index
  int e = 0;
  while (e + 1 < NE && mt >= 4 * c_cumN[e + 1]) ++e;
  const int p = c_P[e];
  const int n = c_N[e];
  const int rowBase = (mt - 4 * c_cumN[e]) * 16;
  const size_t outOff = (size_t)c_cumN[e] * 32768u;  // 64*n*512 per expert
  const int Kpad = (p + 3) & ~3;

  // --- stage A tile + per-row gate scale ---
  {
    const int m  = threadIdx.x >> 4;   // row in tile
    const int t0 = threadIdx.x & 15;
    const int R  = rowBase + m;
    const int nidx = R % n;
    const int bc   = R / n;
    const int ch = bc & 7;
    const int bb = bc >> 3;
    if (t0 == 0)
      lScale[m] = (gates[bb * NE + e] > 0.0f) ? 1.0f : 0.0f;
    for (int k = t0; k < Kpad; k += 16) {
      float v = 0.0f;
      if (k < p) {
        int tpos = nidx * p + k;
        if (tpos > 511) tpos = 511;  // ReplicationPad1d
        v = x[(bb * 512 + tpos) * 8 + ch];
      }
      lA[m * Kpad + k] = v;
    }
  }
  __syncthreads();

  const int wid  = threadIdx.x >> 5;  // 0..7
  const int lane = threadIdx.x & 31;
  const int hl = lane >> 4;
  const int mr = lane & 15;
  const int nc0 = (wid +  0) * 16 + mr;
  const int nc1 = (wid +  8) * 16 + mr;
  const int nc2 = (wid + 16) * 16 + mr;
  const int nc3 = (wid + 24) * 16 + mr;
  const float* W = ptrs.w[e];

  v8f a0 = {}, a1 = {}, a2 = {}, a3 = {};
  for (int k0 = 0; k0 < Kpad; k0 += 4) {
    const int ka = k0 + hl * 2;
    v2f af;
    af.x = lA[mr * Kpad + ka];
    af.y = lA[mr * Kpad + ka + 1];
    const bool v0 = ka < p, v1 = (ka + 1) < p;
    v2f b0, b1_, b2_, b3_;
    b0.x  = v0 ? W[ka * 512 + nc0] : 0.0f;  b0.y  = v1 ? W[(ka + 1) * 512 + nc0] : 0.0f;
    b1_.x = v0 ? W[ka * 512 + nc1] : 0.0f;  b1_.y = v1 ? W[(ka + 1) * 512 + nc1] : 0.0f;
    b2_.x = v0 ? W[ka * 512 + nc2] : 0.0f;  b2_.y = v1 ? W[(ka + 1) * 512 + nc2] : 0.0f;
    b3_.x = v0 ? W[ka * 512 + nc3] : 0.0f;  b3_.y = v1 ? W[(ka + 1) * 512 + nc3] : 0.0f;
    a0 = wmma_f32(af, b0, a0);
    a1 = wmma_f32(af, b1_, a1);
    a2 = wmma_f32(af, b2_, a2);
    a3 = wmma_f32(af, b3_, a3);
  }

#pragma unroll
  for (int j = 0; j < 8; ++j) {
    const int m2 = j + hl * 8;
    const float s = lScale[m2];
    const size_t rbase = outOff + (size_t)(rowBase + m2) * 512;
    out[rbase + nc0] = a0[j] * s;
    out[rbase + nc1] = a1[j] * s;
    out[rbase + nc2] = a2[j] * s;
    out[rbase + nc3] = a3[j] * s;
  }
}

// ---------------------------------------------------------------------------
extern "C" void kernel_launch(void* const* d_in, const int* in_sizes, int n_in,
                              void* d_out, int out_size, void* d_ws, size_t ws_size,
                              hipStream_t stream) {
  (void)in_sizes; (void)n_in; (void)ws_size;
  const float* x  = (const float*)d_in[0];
  const float* w1 = (const float*)d_in[1];
  const float* b1 = (const float*)d_in[2];
  const float* w2 = (const float*)d_in[3];
  const float* b2 = (const float*)d_in[4];
  ExpertPtrs ep;
  for (int i = 0; i < NE; ++i) ep.w[i] = (const float*)d_in[5 + i];

  float* ws  = (float*)d_ws;
  float* out = (float*)d_out;
  const size_t gbase = (size_t)out_size - (size_t)(8 * NE);  // gates tail

  dft_kernel<<<64, 256, 0, stream>>>(x, ws);
  mlp1_kernel<<<32, 256, 0, stream>>>(ws, w1, b1);
  mlp2_kernel<<<dim3(4, 3), 32, 0, stream>>>(ws, w2, b2);
  gates_kernel<<<1, 64, 0, stream>>>(ws, out + gbase);
  embed_kernel<<<6628, 256, 0, stream>>>(x, out + gbase, out, ep);
}